// HausdorffDTLoss_63728724738823
// MI455X (gfx1250) — compile-verified
//
#include <hip/hip_runtime.h>
#include <math.h>

// Problem constants (shape fixed by reference: (16, 1, 320, 320) fp32)
#define NB   16
#define HH   320
#define WW   320
#define HWP  (HH * WW)
#define NPIX (NB * HH * WW)
#define BIGV 1.0e12f
#define FLT_BIG 3.0e38f

#ifndef __has_builtin
#define __has_builtin(x) 0
#endif

// gfx1250 async global->LDS path (ASYNCcnt-tracked), guarded so the file
// still compiles (with plain loads) if the toolchain lacks the builtin.
#if defined(__HIP_DEVICE_COMPILE__) && __has_builtin(__builtin_amdgcn_global_load_async_to_lds_b32)
#define USE_ASYNC_LDS 1
#else
#define USE_ASYNC_LDS 0
#endif

typedef __attribute__((ext_vector_type(2))) float v2f;
typedef __attribute__((ext_vector_type(8))) float v8f;

#if USE_ASYNC_LDS
typedef __attribute__((address_space(1))) int* gas_i32p;   // prints as "__device__ int *"
typedef __attribute__((address_space(3))) int* las_i32p;   // LDS int pointer
#endif

// ---------------------------------------------------------------------------
// Pass 1: 1-D squared distance transform along W for both mask polarities.
// One 320-thread block per (tensor, batch*row). Seeds are 0 / BIG, so the
// brute-force min_j (seed[j] + (i-j)^2) reproduces the reference's min-plus
// bit-for-bit (fp32 add of BIG + exact small int, fp32 min). LDS mask reads
// are same-address broadcasts (conflict-free). Also records has-fg flags.
// ---------------------------------------------------------------------------
__global__ void __launch_bounds__(WW)
dt_row_kernel(const float* __restrict__ pred,
              const float* __restrict__ targ,
              float* __restrict__ fields,   // 4 * NPIX floats
              int* __restrict__ flags)      // 2 * NB ints (pred, target)
{
    __shared__ int mrow[WW];
    __shared__ int anyflag;

    const int t   = blockIdx.y;            // 0 = pred, 1 = target
    const int row = blockIdx.x;            // 0 .. NB*HH-1
    const int b   = row / HH;
    const int x   = threadIdx.x;           // 0 .. 319

    const float* img = (t == 0) ? pred : targ;

    if (x == 0) anyflag = 0;
    __syncthreads();

    const float v = img[row * WW + x];
    const int   m = (v > 0.5f) ? 1 : 0;
    mrow[x] = m;
    if (m) anyflag = 1;                    // benign race: all writers store 1
    __syncthreads();

    if (x == 0 && anyflag) atomicOr(&flags[t * NB + b], 1);

    float mn_fg = FLT_BIG;
    float mn_bg = FLT_BIG;
#pragma unroll 8
    for (int j = 0; j < WW; ++j) {
        const int   dj  = x - j;
        const float d2  = (float)(dj * dj);
        const float sfg = mrow[j] ? BIGV : 0.0f;   // fg seed: mask ? BIG : 0
        const float sbg = BIGV - sfg;              // bg seed: mask ? 0 : BIG
        mn_fg = fminf(mn_fg, sfg + d2);
        mn_bg = fminf(mn_bg, sbg + d2);
    }

    const int px = row * WW + x;
    fields[(t * 2 + 0) * NPIX + px] = mn_fg;
    fields[(t * 2 + 1) * NPIX + px] = mn_bg;
}

// ---------------------------------------------------------------------------
// Pass 2: 1-D min-plus transform along H, in place. Each block owns a
// 320x32 column tile of one field array, staged in LDS (40 KB of the WGP's
// 320 KB) — via GLOBAL_LOAD_ASYNC_TO_LDS_B32 (ASYNCcnt path) when available.
// Loop nest is j-outer with 40 register accumulators per thread: each LDS
// element is read ONCE (320 ds_loads/thread) while the 12,800 min+add terms
// run on VALU out of a register. LDS reads are conflict-free (lane = bank).
// ---------------------------------------------------------------------------
__global__ void __launch_bounds__(256)
dt_col_kernel(float* __restrict__ fields)
{
    __shared__ float tile[HH * 32];

    const int tx  = threadIdx.x;           // 0 .. 31  (column within tile)
    const int ty  = threadIdx.y;           // 0 .. 7
    const int xt  = blockIdx.x % (WW / 32);
    const int b   = blockIdx.x / (WW / 32);
    const int arr = blockIdx.y;            // 0 .. 3 : (tensor*2 + polarity)

    float* f = fields + (size_t)arr * NPIX + (size_t)b * HWP + xt * 32;

#if USE_ASYNC_LDS
    for (int r = ty; r < HH; r += 8) {
        gas_i32p gsrc = (gas_i32p)(f + r * WW + tx);
        las_i32p ldst = (las_i32p)&tile[r * 32 + tx];
        __builtin_amdgcn_global_load_async_to_lds_b32(gsrc, ldst, 0, 0);
    }
#if __has_builtin(__builtin_amdgcn_s_wait_asynccnt)
    __builtin_amdgcn_s_wait_asynccnt(0);
#else
    asm volatile("s_wait_asynccnt 0x0" ::: "memory");
#endif
#else
    for (int r = ty; r < HH; r += 8)
        tile[r * 32 + tx] = f[r * WW + tx];  // coalesced 128B per wave row
#endif
    __syncthreads();

    float mn[HH / 8];
#pragma unroll
    for (int k = 0; k < HH / 8; ++k) mn[k] = FLT_BIG;

    for (int j = 0; j < HH; ++j) {
        const float fv = tile[j * 32 + tx];  // one LDS read feeds 40 terms
#pragma unroll
        for (int k = 0; k < HH / 8; ++k) {
            const int dj = (ty + 8 * k) - j;
            mn[k] = fminf(mn[k], fv + (float)(dj * dj));
        }
    }

#pragma unroll
    for (int k = 0; k < HH / 8; ++k)
        f[(ty + 8 * k) * WW + tx] = mn[k];   // in place: inputs live in LDS
}

// ---------------------------------------------------------------------------
// Pass 3: fused loss + reduction.
//   term = (p-t)^2 * ( [sqrt(min(ffg,BIG)) + sqrt(min(fbg,BIG))]^2  gated by
//                      has_fg, for pred and target )
// Per-wave reduction uses V_WMMA_F32_16X16X4_F32 with B == ones:
//   D[m,n] = acc[m] + acc[m+16]  for every n  (A holds the 32 lane partials)
// Each lane sums its 8 accumulator VGPRs (8 rows of one column) and a single
// shfl_xor(16) folds rows 0-7 with rows 8-15 -> full wave sum.
// ---------------------------------------------------------------------------
__global__ void __launch_bounds__(256)
loss_reduce_kernel(const float* __restrict__ pred,
                   const float* __restrict__ targ,
                   const float* __restrict__ fields,
                   const int* __restrict__ flags,
                   float* __restrict__ out)
{
    const float* __restrict__ fpg = fields + (size_t)0 * NPIX;
    const float* __restrict__ fpb = fields + (size_t)1 * NPIX;
    const float* __restrict__ ftg = fields + (size_t)2 * NPIX;
    const float* __restrict__ ftb = fields + (size_t)3 * NPIX;

    float acc = 0.0f;
    const int tid    = blockIdx.x * blockDim.x + threadIdx.x;
    const int stride = gridDim.x * blockDim.x;

    for (int i = tid; i < NPIX; i += stride) {
        // speculative prefetch of next grid-stride element (global_prefetch_b8)
        __builtin_prefetch(&fpg[i + stride], 0, 1);
        __builtin_prefetch(&ftg[i + stride], 0, 1);

        const int   b = i / HWP;
        const float p = pred[i];
        const float t = targ[i];
        float e = p - t;
        e *= e;

        const float sp = sqrtf(fminf(fpg[i], BIGV)) + sqrtf(fminf(fpb[i], BIGV));
        const float st = sqrtf(fminf(ftg[i], BIGV)) + sqrtf(fminf(ftb[i], BIGV));
        const float dp = flags[b]      ? sp * sp : 0.0f;
        const float dt = flags[NB + b] ? st * st : 0.0f;

        acc += e * (dp + dt);
    }

    // --- wave32 reduction in one matrix op (EXEC is all-ones here) ---
    v2f a;  a.x = acc;   a.y = 0.0f;       // A: 16x4, lanes stripe (m, k-pair)
    v2f bb; bb.x = 1.0f; bb.y = 1.0f;      // B: 4x16 of ones
    v8f c = {};
    v8f d = __builtin_amdgcn_wmma_f32_16x16x4_f32(
        /*neg_a=*/false, a, /*neg_b=*/false, bb,
        /*c_mod=*/(short)0, c, /*reuse_a=*/false, /*reuse_b=*/false);

    float s = d[0] + d[1] + d[2] + d[3] + d[4] + d[5] + d[6] + d[7];
    s += __shfl_xor(s, 16, 32);            // fold rows 0-7 with rows 8-15

    __shared__ float wsum[8];
    const int lane = threadIdx.x & 31;
    const int wv   = threadIdx.x >> 5;
    if (lane == 0) wsum[wv] = s;
    __syncthreads();

    if (threadIdx.x == 0) {
        float bs = 0.0f;
#pragma unroll
        for (int w = 0; w < 8; ++w) bs += wsum[w];
        atomicAdd(out, bs * (1.0f / (float)NPIX));   // mean
    }
}

// ---------------------------------------------------------------------------
extern "C" void kernel_launch(void* const* d_in, const int* in_sizes, int n_in,
                              void* d_out, int out_size, void* d_ws, size_t ws_size,
                              hipStream_t stream)
{
    (void)in_sizes; (void)n_in; (void)out_size; (void)ws_size;

    const float* pred = (const float*)d_in[0];
    const float* targ = (const float*)d_in[1];

    float* fields = (float*)d_ws;                                   // 4*NPIX f32
    int*   flags  = (int*)((char*)d_ws + (size_t)4 * NPIX * sizeof(float));
    float* out    = (float*)d_out;

    (void)hipMemsetAsync(flags, 0, 2 * NB * sizeof(int), stream);
    (void)hipMemsetAsync(out, 0, sizeof(float), stream);

    // Pass 1: W-direction DT (both tensors, both polarities), + has_fg flags
    dim3 g1(NB * HH, 2);
    dt_row_kernel<<<g1, WW, 0, stream>>>(pred, targ, fields, flags);

    // Pass 2: H-direction DT, in place, per 32-column tile
    dim3 g2(NB * (WW / 32), 4);
    dt_col_kernel<<<g2, dim3(32, 8), 0, stream>>>(fields);

    // Pass 3: fused loss + WMMA wave reduction -> scalar mean
    loss_reduce_kernel<<<512, 256, 0, stream>>>(pred, targ, fields, flags, out);
}